// BilateralLayer_35545149342083
// MI455X (gfx1250) — compile-verified
//
#include <hip/hip_runtime.h>

typedef float v2f __attribute__((ext_vector_type(2)));
typedef float v8f __attribute__((ext_vector_type(8)));
typedef unsigned int u32x4 __attribute__((ext_vector_type(4)));
typedef int i32x4 __attribute__((ext_vector_type(4)));
typedef int i32x8 __attribute__((ext_vector_type(8)));

#define BATCH 2
#define HH 1024
#define WW 1024
#define RAD 16

#define TW 32            // workgroup output tile (32x32)
#define CEXT 64          // tile + 2*RAD halo
#define CP 65            // channel tile pitch (floats, odd -> conflict-free cols)
#define MP 17            // mid scratch pitch
#define MIDROWS 48
#define NTHREADS 128
#define NWAVES 4

#define NCH1 25
#define NCH2 16
#define MOMP 33          // moment plane pitch

#if defined(__HIP_DEVICE_COMPILE__) && __has_builtin(__builtin_amdgcn_tensor_load_to_lds)
#define HAVE_TDM 1
#else
#define HAVE_TDM 0
#endif

// 1-D Gaussian tap, d in [-16,16]; theta = R/4 = 4 -> exp(-d^2/32)
__device__ __forceinline__ float gw(int d) {
  return __expf(-(float)(d * d) * (1.0f / 32.0f));
}

// Per-lane banded-Toeplitz A fragments for 16-output / K=48 conv window.
// ISA 16x4 f32 A layout: M = lane%16, K = v + 2*(lane/16); step s covers K=4s..4s+3.
__device__ __forceinline__ void build_amat(v2f* amat, int lane) {
  const int m = lane & 15, h = lane >> 4;
#pragma unroll
  for (int s = 0; s < 12; ++s) {
    const int j0 = 4 * s + 2 * h;
    const int t0 = j0 - m, t1 = j0 + 1 - m;      // tap index in [0,32]
    v2f a;
    a.x = (t0 >= 0 && t0 <= 32) ? gw(t0 - RAD) : 0.0f;
    a.y = (t1 >= 0 && t1 <= 32) ? gw(t1 - RAD) : 0.0f;
    amat[s] = a;
  }
}

// Wave-cooperative 33x33 separable conv on a 16x16 output tile via WMMA f32 16x16x4.
__device__ __forceinline__ v8f conv2d_wave(const float* __restrict__ chan,
                                           float* __restrict__ mid,
                                           const v2f* amat, int wy, int wx,
                                           int lane) {
  const int n = lane & 15;
  const int h = lane >> 4;
  // Horizontal pass: D[M=x_out, N=y] -> mid[48][16]
#pragma unroll
  for (int t = 0; t < 3; ++t) {
    v8f acc = {};
    const int chY = wy + 16 * t + n;
#pragma unroll
    for (int s = 0; s < 12; ++s) {
      const int chX = wx + 4 * s + 2 * h;
      v2f b;
      b.x = chan[chY * CP + chX];
      b.y = chan[chY * CP + chX + 1];
      acc = __builtin_amdgcn_wmma_f32_16x16x4_f32(false, amat[s], false, b,
                                                  (short)0, acc, false, false);
    }
#pragma unroll
    for (int r = 0; r < 8; ++r)
      mid[(16 * t + n) * MP + r + 8 * h] = acc[r];  // mid[y][x]
  }
  // Vertical pass: D[M=y_out, N=x], B rows = mid rows
  v8f acc = {};
#pragma unroll
  for (int s = 0; s < 12; ++s) {
    const int j = 4 * s + 2 * h;
    v2f b;
    b.x = mid[j * MP + n];
    b.y = mid[(j + 1) * MP + n];
    acc = __builtin_amdgcn_wmma_f32_16x16x4_f32(false, amat[s], false, b,
                                                (short)0, acc, false, false);
  }
  return acc;
}

__device__ __forceinline__ void row_col_sums(float* sS, int tid, int gy0, int gx0) {
  if (tid < 64) {
    const int idx = (tid < 32) ? (gy0 + tid) : (gx0 + (tid - 32));
    const int lim = (tid < 32) ? HH : WW;
    float s = 0.0f;
    for (int d = -RAD; d <= RAD; ++d) {
      const int q = idx + d;
      if (q >= 0 && q < lim) s += gw(d);
    }
    sS[tid] = s;
  }
}

#if HAVE_TDM
// TDM: DMA a 64x64 fp32 tile (row stride `stride_elts`) into LDS at byte
// offset `lds_byte`, inserting 1 DWORD of padding every 64 DWORDs (pitch 65).
// Reads past rem_w/rem_h (remaining tensor extent from tile origin) fill zero.
__device__ __forceinline__ void tdm_load_tile_2d(const float* gsrc,
                                                 unsigned lds_byte, int rem_w,
                                                 int rem_h, int stride_elts) {
  const unsigned long long ga = (unsigned long long)(const void*)gsrc;
  const u32x4 g0 = {
      1u,                                            // count=1, load, user mode
      lds_byte,                                      // lds_addr
      (unsigned)(ga & 0xFFFFFFFFu),                  // global_addr[31:0]
      (unsigned)((ga >> 32) & 0x01FFFFFFu) | (2u << 30)  // ga[56:32] | type=2
  };
  const i32x8 g1 = {
      (2 << 16) | (1 << 20) | (5 << 22),   // data_size=4B, pad_en, every 64dw, +1dw
      (rem_w & 0xFFFF) << 16,                          // tensor_dim0[15:0]
      ((rem_w >> 16) & 0xFFFF) | ((rem_h & 0xFFFF) << 16),
      ((rem_h >> 16) & 0xFFFF) | (CEXT << 16),         // tile_dim0=64
      CEXT,                                            // tile_dim1=64, tile_dim2=0
      stride_elts,                                     // tensor_dim0_stride lo
      0,
      0};
  const i32x4 gz = {0, 0, 0, 0};
#if defined(__clang_major__) && __clang_major__ >= 23
  const i32x8 gz8 = {0, 0, 0, 0, 0, 0, 0, 0};
  __builtin_amdgcn_tensor_load_to_lds(g0, g1, gz, gz, gz8, 0);
#else
  __builtin_amdgcn_tensor_load_to_lds(g0, g1, gz, gz, 0);
#endif
}
#endif

__global__ void __launch_bounds__(NTHREADS)
guided_stage1(const float* __restrict__ I, const float* __restrict__ p,
              float* __restrict__ ab) {
  extern __shared__ float smem[];
  float* sS   = smem;                                 // 64
  float* chan = sS + 64;                              // CEXT*CP
  float* midA = chan + CEXT * CP;                     // NWAVES*MIDROWS*MP
  float* mom  = midA + NWAVES * MIDROWS * MP;         // NCH1*TW*MOMP

  const int tid = threadIdx.x;
  const int lane = tid & 31;
  const int wave = tid >> 5;
  const int wy = (wave >> 1) * 16;
  const int wx = (wave & 1) * 16;
  const int bz = blockIdx.z;
  const int gy0 = blockIdx.y * TW;
  const int gx0 = blockIdx.x * TW;
  float* mid = midA + wave * (MIDROWS * MP);

  v2f amat[12];
  build_amat(amat, lane);
  row_col_sums(sS, tid, gy0, gx0);

  const long long planeAll = (long long)BATCH * HH * WW;
  __builtin_prefetch(I + (((long long)bz * HH + gy0) * WW + gx0) * 3, 0, 1);

  for (int ch = 0; ch < NCH1; ++ch) {
    __syncthreads();  // previous iteration done reading chan
    for (int i = tid; i < CEXT * CEXT; i += NTHREADS) {
      const int cy = i >> 6, cx = i & 63;
      const int gy = gy0 - RAD + cy, gx = gx0 - RAD + cx;
      float v = 0.0f;
      if (gy >= 0 && gy < HH && gx >= 0 && gx < WW) {
        const long long base = ((long long)bz * HH + gy) * WW + gx;
        if (ch < 3) {
          v = I[base * 3 + ch];
        } else if (ch < 7) {
          v = p[base * 4 + (ch - 3)];
        } else if (ch < 19) {
          const int e = ch - 7;                       // c = e>>2, k = e&3
          v = I[base * 3 + (e >> 2)] * p[base * 4 + (e & 3)];
        } else {
          const int e = ch - 19;                      // (0,0)(0,1)(0,2)(1,1)(1,2)(2,2)
          const int c1 = (e >= 3) + (e >= 5);
          const int c2 = (e < 3) ? e : ((e < 5) ? (e - 2) : 2);
          v = I[base * 3 + c1] * I[base * 3 + c2];
        }
      }
      chan[cy * CP + cx] = v;
    }
    __syncthreads();
    v8f d = conv2d_wave(chan, mid, amat, wy, wx, lane);
    const int n = lane & 15, h = lane >> 4;
#pragma unroll
    for (int r = 0; r < 8; ++r) {
      const int ly = wy + r + 8 * h, lx = wx + n;
      const float g = d[r] - chan[(ly + RAD) * CP + (lx + RAD)];  // zero center tap
      mom[ch * (TW * MOMP) + ly * MOMP + lx] = g;
    }
  }
  __syncthreads();

  // Per-pixel 3x3 regularized solve -> a (12 planes), b (4 planes)
  for (int pid = tid; pid < TW * TW; pid += NTHREADS) {
    const int ly = pid >> 5, lx = pid & 31;
    const float invN = 1.0f / (sS[ly] * sS[32 + lx] - 1.0f);
    float m[NCH1];
#pragma unroll
    for (int c = 0; c < NCH1; ++c)
      m[c] = mom[c * (TW * MOMP) + ly * MOMP + lx] * invN;

    const float mI0 = m[0], mI1 = m[1], mI2 = m[2];
    const float mp[4] = {m[3], m[4], m[5], m[6]};
    const float v00 = m[19] - mI0 * mI0 + 0.01f;
    const float v01 = m[20] - mI0 * mI1;
    const float v02 = m[21] - mI0 * mI2;
    const float v11 = m[22] - mI1 * mI1 + 0.01f;
    const float v12 = m[23] - mI1 * mI2;
    const float v22 = m[24] - mI2 * mI2 + 0.01f;
    float c00 = v11 * v22 - v12 * v12;
    float c01 = v02 * v12 - v01 * v22;
    float c02 = v01 * v12 - v02 * v11;
    float c11 = v00 * v22 - v02 * v02;
    float c12 = v01 * v02 - v00 * v12;
    float c22 = v00 * v11 - v01 * v01;
    const float id = 1.0f / (v00 * c00 + v01 * c01 + v02 * c02);
    c00 *= id; c01 *= id; c02 *= id; c11 *= id; c12 *= id; c22 *= id;

    const long long pix = ((long long)bz * HH + (gy0 + ly)) * WW + (gx0 + lx);
#pragma unroll
    for (int k = 0; k < 4; ++k) {
      const float cv0 = m[7 + k]     - mI0 * mp[k];
      const float cv1 = m[7 + 4 + k] - mI1 * mp[k];
      const float cv2 = m[7 + 8 + k] - mI2 * mp[k];
      const float a0 = c00 * cv0 + c01 * cv1 + c02 * cv2;
      const float a1 = c01 * cv0 + c11 * cv1 + c12 * cv2;
      const float a2 = c02 * cv0 + c12 * cv1 + c22 * cv2;
      ab[(0 * 4 + k) * planeAll + pix] = a0;
      ab[(1 * 4 + k) * planeAll + pix] = a1;
      ab[(2 * 4 + k) * planeAll + pix] = a2;
      ab[(12 + k) * planeAll + pix] = mp[k] - (a0 * mI0 + a1 * mI1 + a2 * mI2);
    }
  }
}

__global__ void __launch_bounds__(NTHREADS)
guided_stage2(const float* __restrict__ I, const float* __restrict__ ab,
              float* __restrict__ out) {
  extern __shared__ float smem[];
  float* sS   = smem;
  float* chan = sS + 64;
  float* midA = chan + CEXT * CP;
  float* mom  = midA + NWAVES * MIDROWS * MP;         // NCH2*TW*MOMP

  const int tid = threadIdx.x;
  const int lane = tid & 31;
  const int wave = tid >> 5;
  const int wy = (wave >> 1) * 16;
  const int wx = (wave & 1) * 16;
  const int bz = blockIdx.z;
  const int gy0 = blockIdx.y * TW;
  const int gx0 = blockIdx.x * TW;
  float* mid = midA + wave * (MIDROWS * MP);

  v2f amat[12];
  build_amat(amat, lane);
  row_col_sums(sS, tid, gy0, gx0);

  const long long planeAll = (long long)BATCH * HH * WW;
  __builtin_prefetch(ab + ((long long)bz * HH + gy0) * WW + gx0, 0, 1);

  // Interior blocks: top/left halo in range; right/bottom handled by TDM OOB zero-fill.
  const bool interior = (blockIdx.x > 0) && (blockIdx.y > 0);
#if HAVE_TDM
  const unsigned chan_lds = (unsigned)(unsigned long long)(const void*)chan;
#endif

  for (int ch = 0; ch < NCH2; ++ch) {
    __syncthreads();
    const float* src = ab + (long long)ch * planeAll;
#if HAVE_TDM
    if (interior) {
      if (wave == 0) {
        const float* tile0 =
            src + ((long long)bz * HH + (gy0 - RAD)) * WW + (gx0 - RAD);
        tdm_load_tile_2d(tile0, chan_lds, WW - (gx0 - RAD), HH - (gy0 - RAD), WW);
        __builtin_amdgcn_s_wait_tensorcnt(0);
      }
    } else
#else
    (void)interior;
#endif
    {
      for (int i = tid; i < CEXT * CEXT; i += NTHREADS) {
        const int cy = i >> 6, cx = i & 63;
        const int gy = gy0 - RAD + cy, gx = gx0 - RAD + cx;
        float v = 0.0f;
        if (gy >= 0 && gy < HH && gx >= 0 && gx < WW)
          v = src[((long long)bz * HH + gy) * WW + gx];
        chan[cy * CP + cx] = v;
      }
    }
    __syncthreads();
    v8f d = conv2d_wave(chan, mid, amat, wy, wx, lane);
    const int n = lane & 15, h = lane >> 4;
#pragma unroll
    for (int r = 0; r < 8; ++r) {
      const int ly = wy + r + 8 * h, lx = wx + n;
      const float g = d[r] - chan[(ly + RAD) * CP + (lx + RAD)];
      mom[ch * (TW * MOMP) + ly * MOMP + lx] = g;
    }
  }
  __syncthreads();

  for (int pid = tid; pid < TW * TW; pid += NTHREADS) {
    const int ly = pid >> 5, lx = pid & 31;
    const float invN = 1.0f / (sS[ly] * sS[32 + lx] - 1.0f);
    const long long pix = ((long long)bz * HH + (gy0 + ly)) * WW + (gx0 + lx);
    const float I0 = I[pix * 3 + 0];
    const float I1 = I[pix * 3 + 1];
    const float I2 = I[pix * 3 + 2];
#pragma unroll
    for (int k = 0; k < 4; ++k) {
      const float ma0 = mom[(0 * 4 + k) * (TW * MOMP) + ly * MOMP + lx] * invN;
      const float ma1 = mom[(1 * 4 + k) * (TW * MOMP) + ly * MOMP + lx] * invN;
      const float ma2 = mom[(2 * 4 + k) * (TW * MOMP) + ly * MOMP + lx] * invN;
      const float mb  = mom[(12 + k)    * (TW * MOMP) + ly * MOMP + lx] * invN;
      out[pix * 4 + k] = ma0 * I0 + ma1 * I1 + ma2 * I2 + mb;
    }
  }
}

extern "C" void kernel_launch(void* const* d_in, const int* in_sizes, int n_in,
                              void* d_out, int out_size, void* d_ws, size_t ws_size,
                              hipStream_t stream) {
  const float* I = (const float*)d_in[0];
  const float* p = (const float*)d_in[1];
  float* ab = (float*)d_ws;       // 16 planes * B*H*W fp32 = 128 MB
  float* out = (float*)d_out;

  dim3 grid(WW / TW, HH / TW, BATCH);
  dim3 block(NTHREADS);
  const size_t smem1 =
      (size_t)(64 + CEXT * CP + NWAVES * MIDROWS * MP + NCH1 * TW * MOMP) * sizeof(float);
  const size_t smem2 =
      (size_t)(64 + CEXT * CP + NWAVES * MIDROWS * MP + NCH2 * TW * MOMP) * sizeof(float);

  guided_stage1<<<grid, block, smem1, stream>>>(I, p, ab);
  guided_stage2<<<grid, block, smem2, stream>>>(I, ab, out);
}